// CLUBLoss_10651518894749
// MI455X (gfx1250) — compile-verified
//
#include <hip/hip_runtime.h>
#include <hip/hip_bf16.h>

// CDNA5 / gfx1250 — wave32, WMMA f32 16x16x4 used as a row-sum engine.
typedef __attribute__((ext_vector_type(2))) float v2f;
typedef __attribute__((ext_vector_type(8))) float v8f;

#define B_ 16
#define D_ 64
#define HW_ 1024            // 32*32
#define N_ (B_ * HW_)       // 16384
#define ROWS_PER_BLOCK 128  // 8 waves x 16 rows
#define NBLK (N_ / ROWS_PER_BLOCK)  // 128

// ---------------- Kernel 1: per-channel mean(x) and mean(x^2) ----------------
__global__ void club_stats_kernel(const float* __restrict__ x,
                                  float* __restrict__ mx,
                                  float* __restrict__ mx2) {
    const int dd = blockIdx.x;          // channel, 64 blocks
    const int t  = threadIdx.x;         // 256 threads
    float s = 0.f, s2 = 0.f;
    for (int b = 0; b < B_; ++b) {
        const float* base = x + (size_t)b * (D_ * HW_) + (size_t)dd * HW_;
        for (int i = t; i < HW_; i += 256) {   // fully coalesced
            float v = base[i];
            s += v;
            s2 += v * v;
        }
    }
    __shared__ float rs[256];
    __shared__ float rq[256];
    rs[t] = s; rq[t] = s2;
    __syncthreads();
    for (int off = 128; off > 0; off >>= 1) {
        if (t < off) { rs[t] += rs[t + off]; rq[t] += rq[t + off]; }
        __syncthreads();
    }
    if (t == 0) {
        const float invN = 1.0f / (float)N_;
        mx[dd]  = rs[0] * invN;
        mx2[dd] = rq[0] * invN;
    }
}

// ---------------- Kernel 2: main streaming pass + WMMA row-sum ----------------
// loss term per (i,d): t = inv_v * ( mx2 - 2*mu*mx + mu^2 - (x-mu)^2 )
// Each wave: 16 rows; 16 K-steps of V_WMMA_F32_16X16X4_F32 with B = ones
// accumulate row sums into the f32 C tile (each row sum duplicated across 16 lanes).
__global__ void club_main_kernel(const float* __restrict__ x,
                                 const float* __restrict__ mu,
                                 const float* __restrict__ logvar,
                                 const float* __restrict__ mx,
                                 const float* __restrict__ mx2,
                                 float* __restrict__ partials) {
    const int tid  = threadIdx.x;
    const int lane = tid & 31;
    const int wave = tid >> 5;                       // 8 waves
    const int m    = lane & 15;                      // row within 16-row tile
    const int khi  = lane >> 4;                      // 0 -> K{0,1}, 1 -> K{2,3}

    const int gr = blockIdx.x * ROWS_PER_BLOCK + wave * 16 + m;  // global row
    const int b  = gr >> 10;                         // gr / 1024
    const int p  = gr & 1023;                        // plane offset (hh*32+ww)

    const float* __restrict__ xrow  = x + (size_t)b * (D_ * HW_) + p; // + c*HW_
    const float* __restrict__ murow = mu + (size_t)gr * D_;
    const float* __restrict__ lvrow = logvar + (size_t)gr * D_;

    v8f acc = {};                                    // 16x16 f32 C/D tile
    const v2f ones = {1.0f, 1.0f};                   // B = all-ones (4x16)

    #pragma unroll 4
    for (int s = 0; s < 16; ++s) {
        const int c0 = s * 4 + khi * 2;              // this lane's 2 channels
        float2 m2  = *(const float2*)(murow + c0);
        float2 l2  = *(const float2*)(lvrow + c0);
        float2 mxv = *(const float2*)(mx  + c0);
        float2 mqv = *(const float2*)(mx2 + c0);
        float  x0  = xrow[(size_t)c0 * HW_];
        float  x1  = xrow[(size_t)(c0 + 1) * HW_];
        if (s < 15) {                                // warm next K-step (global_prefetch_b8)
            __builtin_prefetch(murow + c0 + 4, 0, 1);
            __builtin_prefetch(lvrow + c0 + 4, 0, 1);
        }
        float iv0 = __expf(-l2.x);
        float iv1 = __expf(-l2.y);
        float dx0 = x0 - m2.x;
        float dx1 = x1 - m2.y;
        float t0 = iv0 * (mqv.x - 2.0f * m2.x * mxv.x + m2.x * m2.x - dx0 * dx0);
        float t1 = iv1 * (mqv.y - 2.0f * m2.y * mxv.y + m2.y * m2.y - dx1 * dx1);
        v2f a = {t0, t1};                            // A-tile 16x4 f32 slice
        // D[i,j] += sum_k A[i,k]*1  -> row sums accumulate in C
        acc = __builtin_amdgcn_wmma_f32_16x16x4_f32(
            /*neg_a=*/false, a, /*neg_b=*/false, ones,
            /*c_mod=*/(short)0, acc, /*reuse_a=*/false, /*reuse_b=*/false);
    }

    // Per-lane: sum the 8 C registers; block tree-reduce (deterministic).
    float sl = acc[0] + acc[1] + acc[2] + acc[3] + acc[4] + acc[5] + acc[6] + acc[7];
    __shared__ float red[256];
    red[tid] = sl;
    __syncthreads();
    for (int off = 128; off > 0; off >>= 1) {
        if (tid < off) red[tid] += red[tid + off];
        __syncthreads();
    }
    if (tid == 0) {
        // each row sum counted 16x (once per N-column lane)
        partials[blockIdx.x] = red[0] * (1.0f / 16.0f);
    }
}

// ---------------- Kernel 3: deterministic final reduction ----------------
__global__ void club_final_kernel(const float* __restrict__ partials,
                                  float* __restrict__ out) {
    const int t = threadIdx.x;                       // 128 threads, 1 block
    __shared__ float r[NBLK];
    r[t] = partials[t];
    __syncthreads();
    for (int off = NBLK / 2; off > 0; off >>= 1) {
        if (t < off) r[t] += r[t + off];
        __syncthreads();
    }
    if (t == 0) out[0] = r[0] * (0.5f / (float)N_);
}

extern "C" void kernel_launch(void* const* d_in, const int* in_sizes, int n_in,
                              void* d_out, int out_size, void* d_ws, size_t ws_size,
                              hipStream_t stream) {
    (void)in_sizes; (void)n_in; (void)out_size; (void)ws_size;
    const float* x      = (const float*)d_in[0];   // (16,64,32,32)
    const float* mu     = (const float*)d_in[1];   // (16384,64)
    const float* logvar = (const float*)d_in[2];   // (16384,64)
    float* out = (float*)d_out;

    float* ws       = (float*)d_ws;
    float* mx       = ws;                 // 64
    float* mx2      = ws + 64;            // 64
    float* partials = ws + 128;           // 128

    club_stats_kernel<<<D_, 256, 0, stream>>>(x, mx, mx2);
    club_main_kernel<<<NBLK, 256, 0, stream>>>(x, mu, logvar, mx, mx2, partials);
    club_final_kernel<<<1, NBLK, 0, stream>>>(partials, out);
}